// RadarRnn5_24558622999004
// MI455X (gfx1250) — compile-verified
//
#include <hip/hip_runtime.h>
#include <hip/hip_bf16.h>

typedef __attribute__((ext_vector_type(2))) float v2f;
typedef __attribute__((ext_vector_type(8))) float v8f;

#define HID     64
#define TLEN    2048
#define BATCH   256
#define NLAYERS 8
#define RP      68      // padded LDS row stride (floats): bank-conflict-free
#define KSTEPS  16      // 64 / 4 (K per f32 WMMA)
#define NTHREADS (NLAYERS * 2 * 32)   // 2 waves per layer

#if __has_builtin(__builtin_amdgcn_tanhf)
#define TANH(x) __builtin_amdgcn_tanhf(x)
#else
#define TANH(x) tanhf(x)
#endif

__global__ __launch_bounds__(NTHREADS, 1)
void rnn_stack_kernel(const float* __restrict__ x,      // [B,T,64]
                      const float* __restrict__ h0,     // [8,B,64]
                      const float* __restrict__ Wih0,   // [64,64]
                      const float* __restrict__ Wih,    // [7,64,64]
                      const float* __restrict__ Whh,    // [8,64,64]
                      const float* __restrict__ bih,    // [8,64]
                      const float* __restrict__ bhh,    // [8,64]
                      const float* __restrict__ Wout,   // [1,64]
                      const float* __restrict__ bout,   // [1]
                      float* __restrict__ y,            // [B,T]
                      float* __restrict__ hfinal)       // [8,B,64]
{
  __shared__ float bufs[NLAYERS][16 * RP];   // 8 x 16x68 f32 tiles = 34 KB
  __shared__ float yscr[2][16];              // top-layer partial row sums

  const int tid  = threadIdx.x;
  // wave id -> (layer, N-half); force into SGPR so pipeline guards are scalar
  // branches (WMMA must never execute under a lane mask).
  const int wv   = __builtin_amdgcn_readfirstlane(tid >> 5);
  const int l    = wv >> 1;          // layer 0..7
  const int w    = wv & 1;           // N-half 0..1 (cols 32w .. 32w+31)
  const int lane = tid & 31;
  const int lo   = lane & 15;        // M (A) or N (B/C) within tile
  const int hi   = lane >> 4;        // half-wave select
  const int b0   = blockIdx.x * 16;  // batch-tile start row

  // ---- per-layer weight fragments -> registers (B-operand layout) ----
  // B(k,n) = W[n][k]  (out = inp @ W.T). Lane: n = (2w+j)*16+lo, K-half = hi.
  const float* WihL = (l == 0) ? Wih0 : (Wih + (size_t)(l - 1) * HID * HID);
  const float* WhhL = Whh + (size_t)l * HID * HID;

  v2f   wih[2][KSTEPS];
  v2f   whh[2][KSTEPS];
  float bias[2];
  float wout[2];

#pragma unroll
  for (int j = 0; j < 2; ++j) {
    const int n = (2 * w + j) * 16 + lo;
    bias[j] = bih[l * HID + n] + bhh[l * HID + n];
    wout[j] = Wout[n];
#pragma unroll
    for (int k = 0; k < KSTEPS; ++k) {
      const int off = n * HID + 4 * k + 2 * hi;
      wih[j][k] = *(const v2f*)(WihL + off);
      whh[j][k] = *(const v2f*)(WhhL + off);
    }
  }
  const float bo = bout[0];

  // ---- seed hidden state tiles from h_state (2 waves per layer) ----
  for (int i = lane + 32 * w; i < 16 * HID; i += 64) {
    const int m = i >> 6;
    const int n = i & (HID - 1);
    bufs[l][m * RP + n] = h0[((size_t)l * BATCH + b0 + m) * HID + n];
  }
  __syncthreads();

  // ---- layer-pipelined recurrent loop: layer l handles t = s - l ----
  for (int s = 0; s < TLEN + NLAYERS; ++s) {
    const int  t      = s - l;
    const bool active = (t >= 0) && (t < TLEN);

    // ---------------- phase 1: reads ----------------
    v2f a_in[KSTEPS], a_h[KSTEPS];
    if (active) {
      // A fragments: lane m=lo, v2f = cols {4k+2hi, 4k+2hi+1}
      if (l == 0) {
        const float* xr = x + ((size_t)(b0 + lo) * TLEN + t) * HID;
#pragma unroll
        for (int k = 0; k < KSTEPS; ++k)
          a_in[k] = *(const v2f*)(xr + 4 * k + 2 * hi);
      } else {
        const float* sr = &bufs[l - 1][lo * RP];
#pragma unroll
        for (int k = 0; k < KSTEPS; ++k)
          a_in[k] = *(const v2f*)(sr + 4 * k + 2 * hi);
      }
      const float* hr = &bufs[l][lo * RP];
#pragma unroll
      for (int k = 0; k < KSTEPS; ++k)
        a_h[k] = *(const v2f*)(hr + 4 * k + 2 * hi);
    }

    // flush y[t-1] for the top layer (partials written last superstep)
    {
      const int ty = s - 8;
      if (l == NLAYERS - 1 && w == 0 && ty >= 0 && ty < TLEN) {
        if (lane < 16)
          y[(size_t)(b0 + lane) * TLEN + ty] = yscr[0][lane] + yscr[1][lane] + bo;
      }
    }
    __syncthreads();   // all reads of superstep s done before any writes

    // ---------------- phase 2: compute + writes ----------------
    if (active) {
      v8f acc[2];
#pragma unroll
      for (int j = 0; j < 2; ++j)
#pragma unroll
        for (int r = 0; r < 8; ++r) acc[j][r] = bias[j];

      // pre += inp @ Wih.T   (2 independent accumulator chains)
#pragma unroll
      for (int k = 0; k < KSTEPS; ++k)
#pragma unroll
        for (int j = 0; j < 2; ++j)
          acc[j] = __builtin_amdgcn_wmma_f32_16x16x4_f32(
              false, a_in[k], false, wih[j][k], (short)0, acc[j], false, false);

      // pre += h @ Whh.T
#pragma unroll
      for (int k = 0; k < KSTEPS; ++k)
#pragma unroll
        for (int j = 0; j < 2; ++j)
          acc[j] = __builtin_amdgcn_wmma_f32_16x16x4_f32(
              false, a_h[k], false, whh[j][k], (short)0, acc[j], false, false);

      // tanh + write h tile to own LDS buffer (C layout -> row-major padded)
      float* dst = bufs[l];
      float  yp[8];
#pragma unroll
      for (int r = 0; r < 8; ++r) yp[r] = 0.0f;

#pragma unroll
      for (int j = 0; j < 2; ++j) {
#pragma unroll
        for (int r = 0; r < 8; ++r) {
          const float hv = TANH(acc[j][r]);
          dst[(r + 8 * hi) * RP + (2 * w + j) * 16 + lo] = hv;
          if (l == NLAYERS - 1) yp[r] += hv * wout[j];
        }
      }

      // top layer: reduce partial output projection, stash in LDS
      if (l == NLAYERS - 1) {
#pragma unroll
        for (int r = 0; r < 8; ++r) {
          float p = yp[r];
          p += __shfl_xor(p, 8);
          p += __shfl_xor(p, 4);
          p += __shfl_xor(p, 2);
          p += __shfl_xor(p, 1);
          if (lo == 0)   // lanes 0 (row r) and 16 (row r+8)
            yscr[w][r + 8 * hi] = p;
        }
      }
    }
    __syncthreads();   // writes of superstep s visible before s+1 reads
  }

  // ---- flush final hidden states ----
  for (int i = lane + 32 * w; i < 16 * HID; i += 64) {
    const int m = i >> 6;
    const int n = i & (HID - 1);
    hfinal[((size_t)l * BATCH + b0 + m) * HID + n] = bufs[l][m * RP + n];
  }
}

extern "C" void kernel_launch(void* const* d_in, const int* in_sizes, int n_in,
                              void* d_out, int out_size, void* d_ws, size_t ws_size,
                              hipStream_t stream) {
  (void)in_sizes; (void)n_in; (void)d_ws; (void)ws_size; (void)out_size;
  const float* x    = (const float*)d_in[0];
  const float* h0   = (const float*)d_in[1];
  const float* Wih0 = (const float*)d_in[2];
  const float* Wih  = (const float*)d_in[3];
  const float* Whh  = (const float*)d_in[4];
  const float* bih  = (const float*)d_in[5];
  const float* bhh  = (const float*)d_in[6];
  const float* Wout = (const float*)d_in[7];
  const float* bout = (const float*)d_in[8];

  float* y  = (float*)d_out;                       // [B,T,1] flat
  float* hf = y + (size_t)BATCH * TLEN;            // [8,B,64] flat

  rnn_stack_kernel<<<BATCH / 16, NTHREADS, 0, stream>>>(
      x, h0, Wih0, Wih, Whh, bih, bhh, Wout, bout, y, hf);
}